// TimeDecayedAveragePoolingLayerWithMask_36532991819826
// MI455X (gfx1250) — compile-verified
//
#include <hip/hip_runtime.h>
#include <stdint.h>

// Time-decayed masked average pooling (B=16384, S=20, D=1024, fp32):
//   out[b,d] = sum_s w[b,s] * x[b,s,d],  w = normalized 0.5^(L-1-s) over first L tokens.
//
// Memory-bound: 1.34 GB streamed once -> ~60us floor @ 23.3 TB/s.
// Pipeline (CDNA5 paths):
//   1) global_load_async_to_lds_b128 stages x[b, 0:20, dhalf] into LDS
//      (512B contiguous per instruction, ASYNCcnt-tracked)
//   2) s_wait_asynccnt 0 + workgroup barrier
//   3) ds_load_b32 gathers in WMMA B-operand layout (bank-conflict-free via row pad)
//   4) V_WMMA_F32_16X16X4_F32 performs the weighted reduction exactly in fp32:
//      B[k,n] = x[b, s0+k, d0+n], A[m,k] = w[s0+k] broadcast -> all D rows == out tile.

typedef __attribute__((ext_vector_type(2))) float v2f;
typedef __attribute__((ext_vector_type(8))) float v8f;

#define POOL_MAXLEN 20
#define POOL_D      1024
#define DHALF       512
#define LDW         528   // 512 + 16 pad: (2*LDW) % 64 == 32 -> half-waves on disjoint LDS banks

__global__ __launch_bounds__(256) void
TimeDecayedAveragePool_wmma_async_kernel(const float* __restrict__ x,
                                         const int* __restrict__ mask,
                                         float* __restrict__ out)
{
    const int bh    = blockIdx.x;
    const int b     = bh >> 1;
    const int dBase = (bh & 1) * DHALF;      // this block handles d in [dBase, dBase+512)
    const int tid   = threadIdx.x;
    const int lane  = tid & 31;
    const int wave  = tid >> 5;              // 0..7, each wave owns 64 d-values of the half

    __shared__ float smem[POOL_MAXLEN * LDW];   // 42,240 bytes

    // ---- 1) async-stage x[b, 0:20, dBase:dBase+512] into LDS (b128 per lane) ----
    const float* __restrict__ xb = x + (size_t)b * (POOL_MAXLEN * POOL_D) + dBase;
    const uint32_t ldsBase = (uint32_t)(uintptr_t)&smem[0];  // flat addr low 32b == LDS offset
    #pragma unroll
    for (int t = 0; t < 10; ++t) {
        const int i   = t * 256 + tid;       // 2560 transfers of 16B = 20 rows x 512 floats
        const int row = i >> 7;              // 128 b128-transfers per row
        const int c4  = (i & 127) << 2;      // starting float column within the half
        const uint32_t gOff = (uint32_t)((row * POOL_D + c4) * 4);
        const uint32_t lOff = ldsBase + (uint32_t)((row * LDW + c4) * 4);
        asm volatile("global_load_async_to_lds_b128 %0, %1, %2"
                     :: "v"(lOff), "v"(gOff), "s"(xb)
                     : "memory");
    }

    // ---- session length L via wave butterfly over mask[b, 0..19] (overlaps async) ----
    int mv = (lane < POOL_MAXLEN) ? mask[b * POOL_MAXLEN + lane] : 0;
    #pragma unroll
    for (int off = 16; off > 0; off >>= 1)
        mv += __shfl_xor(mv, off, 32);
    const int   L       = mv;                                   // uniform across wave
    const float invnorm = 1.0f / (2.0f - __builtin_exp2f((float)(1 - L)));

    // WMMA wave32 32-bit operand layout: lanes 0-15 carry K={0,1}, lanes 16-31 K={2,3};
    // B-matrix column n = lane & 15.
    const int halfSel = lane >> 4;
    const int col     = lane & 15;

    // A operand per K-chunk: broadcast weights w(s), zero past L.  alpha^(L-1-s) = exp2(s-(L-1))
    v2f aw[5];
    #pragma unroll
    for (int c = 0; c < 5; ++c) {
        const int s0 = c * 4 + halfSel * 2;
        const int s1 = s0 + 1;
        aw[c].x = (s0 < L) ? __builtin_exp2f((float)(s0 - (L - 1))) * invnorm : 0.0f;
        aw[c].y = (s1 < L) ? __builtin_exp2f((float)(s1 - (L - 1))) * invnorm : 0.0f;
    }

    // ---- 2) wait own async transfers, then join the workgroup ----
    asm volatile("s_wait_asynccnt 0" ::: "memory");
    __syncthreads();

    // ---- 3+4) 4 tiles of 16 d-values per wave: DS gather -> WMMA ----
    #pragma unroll
    for (int tt = 0; tt < 4; ++tt) {
        const int dL0 = wave * 64 + tt * 16;    // local d within the 512-half

        v8f acc = {};
        #pragma unroll
        for (int c = 0; c < 5; ++c) {
            const int s0 = c * 4 + halfSel * 2;  // K rows this lane carries
            v2f bm;
            bm.x = smem[s0 * LDW + dL0 + col];
            bm.y = smem[(s0 + 1) * LDW + dL0 + col];
            acc = __builtin_amdgcn_wmma_f32_16x16x4_f32(
                /*neg_a=*/false, aw[c],
                /*neg_b=*/false, bm,
                /*c_mod=*/(short)0, acc,
                /*reuse_a=*/false, /*reuse_b=*/false);
        }

        // every D row identical; row M=0 lives in C-VGPR0, lanes 0-15
        if (lane < 16)
            __builtin_nontemporal_store(acc[0],
                out + (size_t)b * POOL_D + dBase + dL0 + lane);
    }
}

extern "C" void kernel_launch(void* const* d_in, const int* in_sizes, int n_in,
                              void* d_out, int out_size, void* d_ws, size_t ws_size,
                              hipStream_t stream)
{
    (void)n_in; (void)d_ws; (void)ws_size; (void)out_size;
    const float* x    = (const float*)d_in[0];
    const int*   mask = (const int*)d_in[1];
    float*       out  = (float*)d_out;

    const int B = in_sizes[1] / POOL_MAXLEN;   // mask is [B, MAXLEN]

    dim3 grid((unsigned)(2 * B));              // one block per (session, 512-wide d half)
    dim3 block(256);
    TimeDecayedAveragePool_wmma_async_kernel<<<grid, block, 0, stream>>>(x, mask, out);
}